// crossCorrelation3D_13546326851852
// MI455X (gfx1250) — compile-verified
//
#include <hip/hip_runtime.h>
#include <hip/hip_bf16.h>

// ---------------------------------------------------------------------------
// LNCC (9x9x9 box window, SAME zero padding) loss on [2,1,160,160,160] f32.
// Separable 2-kernel plan (bandwidth-bound on MI455X, 23.3 TB/s):
//   Kernel A: pointwise fields + W-sum (LDS taps) + H-sum (WMMA banded matmul)
//   Kernel B: D-sum (TDM tensor_load_to_lds -> LDS -> WMMA) + fused finalize
// ---------------------------------------------------------------------------

typedef float v2f  __attribute__((ext_vector_type(2)));
typedef float v8f  __attribute__((ext_vector_type(8)));
typedef unsigned int u32x4 __attribute__((ext_vector_type(4)));
typedef int   i32x8 __attribute__((ext_vector_type(8)));
typedef int   i32x4 __attribute__((ext_vector_type(4)));

#define L        160
#define SLICE    25600          // 160*160
#define NV       8192000UL     // 2*160^3 voxels per volume
#define KS       729.0f

// ---------------------------------------------------------------------------
// Kernel A: per (b,d) slice, 16x16 output tile with 24x24 halo.
//  - load input/target halo (zero-filled OOB -> SAME padding of the fields)
//  - compute 5 fields into LDS
//  - W-direction 9-tap sum -> Sw[24 rows][16]
//  - H-direction sum as D = Band(16x24) x Sw(24x16) via 6x v_wmma_f32_16x16x4_f32
//  - 5 waves, one field each
// ---------------------------------------------------------------------------
__global__ __launch_bounds__(160)
void lncc_wh_pass(const float* __restrict__ input,
                  const float* __restrict__ target,
                  float* __restrict__ ws)
{
    __shared__ float fld[5 * 576];   // 5 fields x 24x24 halo
    __shared__ float sw [5 * 384];   // 5 fields x 24(h) x 16(w) W-summed

    const int tid = threadIdx.x;
    const int w0  = blockIdx.x * 16;
    const int h0  = blockIdx.y * 16;
    const int bd  = blockIdx.z;                  // b*160 + d
    const size_t slice = (size_t)bd * SLICE;

    // ---- load halo + pointwise fields (OOB => 0, i.e. zero padding of fields)
    for (int e = tid; e < 576; e += 160) {
        const int r  = e / 24, c = e % 24;
        const int gh = h0 - 4 + r;
        const int gw = w0 - 4 + c;
        float fT = 0.f, fI = 0.f, fTT = 0.f, fII = 0.f, fIT = 0.f;
        if (gh >= 0 && gh < L && gw >= 0 && gw < L) {
            const size_t gi = slice + (size_t)gh * L + gw;
            const float in = input[gi];
            const float tg = (target[gi] + 1.0f) * 0.5f;
            fT = tg; fI = in; fTT = tg * tg; fII = in * in; fIT = in * tg;
        }
        fld[0*576 + e] = fT;
        fld[1*576 + e] = fI;
        fld[2*576 + e] = fTT;
        fld[3*576 + e] = fII;
        fld[4*576 + e] = fIT;
    }
    __syncthreads();

    // ---- W-direction 9-tap box sum: Sw[f][r][n] = sum_{c=n..n+8} F[f][r][c]
    for (int e = tid; e < 1920; e += 160) {
        const int f = e / 384, rem = e % 384, r = rem / 16, n = rem % 16;
        const float* src = &fld[f*576 + r*24 + n];
        float s = 0.f;
#pragma unroll
        for (int j = 0; j < 9; ++j) s += src[j];
        sw[f*384 + r*16 + n] = s;
    }
    __syncthreads();

    // ---- H-direction sum as banded matmul. Wave f handles field f.
    const int f    = tid >> 5;          // 0..4
    const int lane = tid & 31;
    const int n    = lane & 15;         // D/B column, also A row m
    const int kh   = lane >> 4;         // 0: K even pair base 0, 1: base +2

    v8f acc = {};
#pragma unroll
    for (int ch = 0; ch < 6; ++ch) {
        const int k0 = 4*ch + 2*kh;     // f32 A/B layout: VGPR0 K=k0, VGPR1 K=k0+1
        v2f a, b;
        a.x = (k0     >= n && k0     <= n + 8) ? 1.0f : 0.0f;   // Band[m][k0]
        a.y = (k0 + 1 >= n && k0 + 1 <= n + 8) ? 1.0f : 0.0f;   // Band[m][k0+1]
        b.x = sw[f*384 + (k0    ) * 16 + n];
        b.y = sw[f*384 + (k0 + 1) * 16 + n];
        acc = __builtin_amdgcn_wmma_f32_16x16x4_f32(
                  false, a, false, b, (short)0, acc, false, false);
    }

    // ---- store D: VGPR i holds rows M=i (lanes 0-15) / M=i+8 (lanes 16-31)
#pragma unroll
    for (int i = 0; i < 8; ++i) {
        const int h = h0 + i + 8*kh;
        ws[(size_t)f * NV + slice + (size_t)h * L + w0 + n] = acc[i];
    }
}

// ---------------------------------------------------------------------------
// Kernel B: one wave per 16(d) x 16(w) tile at fixed (b,h).
//  - TDM loads 5 strided 24x16 tiles (stride = 160*160 elems) into LDS
//    (tensor_dim1 zero-fills d>=160; LDS shift + pre-zero handles d<0)
//  - 5 x (6 x v_wmma_f32_16x16x4_f32) banded matmuls for the D-sum
//  - fused finalize: cc formula, sigmoid(flow) weight, wave reduce, atomicAdd
// ---------------------------------------------------------------------------
__global__ __launch_bounds__(32)
void lncc_d_finalize(const float* __restrict__ ws,
                     const float* __restrict__ flow,
                     float* __restrict__ out)
{
    __shared__ float tile[5 * 384];     // 5 fields x 24(d) x 16(w)

    const int lane = threadIdx.x;
    const int w0 = blockIdx.x * 16;
    const int d0 = blockIdx.y * 16;
    const int b  = blockIdx.z / L;
    const int h  = blockIdx.z % L;

    const int shift  = (d0 == 0) ? 4 : 0;        // rows below d=0 are zero pad
    const int dstart = d0 - 4 + shift;           // first real row fetched
    const unsigned td1   = (unsigned)(L - dstart);   // rows left -> OOB=>0 above
    const unsigned tile1 = (unsigned)(24 - shift);   // rows the TDM writes

    // pre-zero the padded leading rows (disjoint from TDM destination)
    if (shift) {
        for (int e = lane; e < 5 * 4 * 16; e += 32) {
            const int f = e / 64;
            tile[f*384 + (e % 64)] = 0.f;
        }
    }

    const unsigned ldsbase = (unsigned)(size_t)(void*)tile; // low 32b = LDS offset

#pragma unroll
    for (int f = 0; f < 5; ++f) {
        const unsigned long long ga = (unsigned long long)(size_t)(const void*)
            (ws + (size_t)f * NV
                + (((size_t)b * L + (size_t)dstart) * SLICE + (size_t)h * L + w0));

        u32x4 g0;
        g0[0] = 1u;                                            // count=1, load D#
        g0[1] = ldsbase + (unsigned)(f * 384 + shift * 16) * 4u; // lds_addr bytes
        g0[2] = (unsigned)(ga & 0xFFFFFFFFull);                // global_addr lo
        g0[3] = (unsigned)((ga >> 32) & 0x1FFFFFFull) | (2u << 30); // addr hi|type=2

        i32x8 g1;
        g1[0] = (int)(2u << 16);                 // data_size=4B
        g1[1] = (int)(16u << 16);                // tensor_dim0 = 16 (bits 63:48)
        g1[2] = (int)((td1 & 0xFFFFu) << 16);    // tensor_dim1 lo16 (bits 95:80)
        g1[3] = (int)((td1 >> 16) | (16u << 16));// tensor_dim1 hi | tile_dim0=16
        g1[4] = (int)(tile1 & 0xFFFFu);          // tile_dim1 | tile_dim2=0
        g1[5] = (int)SLICE;                      // tensor_dim0_stride = 25600 elems
        g1[6] = 0;
        g1[7] = 0;

        i32x4 g2 = {0, 0, 0, 0};                 // 2D tensor: dims 2+ unused
        i32x4 g3 = {0, 0, 0, 0};
        i32x8 g4 = {0, 0, 0, 0, 0, 0, 0, 0};     // extra group (clang-23 6-arg form)

        __builtin_amdgcn_tensor_load_to_lds(g0, g1, g2, g3, g4, 0);
    }
    __builtin_amdgcn_s_wait_tensorcnt(0);

    // ---- D-direction banded matmuls, one accumulator set per field
    const int n  = lane & 15;
    const int kh = lane >> 4;

    v8f acc[5];
#pragma unroll
    for (int f = 0; f < 5; ++f) {
        v8f c = {};
#pragma unroll
        for (int ch = 0; ch < 6; ++ch) {
            const int k0 = 4*ch + 2*kh;
            v2f a, bb;
            a.x = (k0     >= n && k0     <= n + 8) ? 1.0f : 0.0f;
            a.y = (k0 + 1 >= n && k0 + 1 <= n + 8) ? 1.0f : 0.0f;
            bb.x = tile[f*384 + (k0    ) * 16 + n];
            bb.y = tile[f*384 + (k0 + 1) * 16 + n];
            c = __builtin_amdgcn_wmma_f32_16x16x4_f32(
                    false, a, false, bb, (short)0, c, false, false);
        }
        acc[f] = c;
    }

    // ---- fused finalize: each lane owns 8 voxels (rows i / i+8 per D layout)
    float lsum = 0.f;
#pragma unroll
    for (int i = 0; i < 8; ++i) {
        const int d = d0 + i + 8*kh;
        const float Ts = acc[0][i];
        const float Is = acc[1][i];
        const float TT = acc[2][i];
        const float II = acc[3][i];
        const float IT = acc[4][i];

        const float Ihat = Is * (1.0f / KS);
        const float That = Ts * (1.0f / KS);
        const float cross = IT - Ihat * Ts - That * Is + That * Ihat * KS;
        const float Tvar  = TT - 2.0f * That * Ts + That * That * KS;
        const float Ivar  = II - 2.0f * Ihat * Is + Ihat * Ihat * KS;

        const size_t gi = (((size_t)b * L + d) * L + h) * L + w0 + n;
        const float wgt = 1.0f / (1.0f + __expf(-flow[gi]));   // sigmoid^1

        lsum += cross * cross * wgt / (Tvar * Ivar + 1e-5f);
    }

    // wave32 reduction, then one atomic per wave: accumulate -mean contribution
#pragma unroll
    for (int off = 16; off > 0; off >>= 1)
        lsum += __shfl_xor(lsum, off, 32);
    if (lane == 0)
        atomicAdd(out, lsum * (-1.0f / 8192000.0f));
}

// ---------------------------------------------------------------------------
extern "C" void kernel_launch(void* const* d_in, const int* in_sizes, int n_in,
                              void* d_out, int out_size, void* d_ws, size_t ws_size,
                              hipStream_t stream) {
    const float* input  = (const float*)d_in[0];
    const float* target = (const float*)d_in[1];
    const float* flow   = (const float*)d_in[2];
    float* out = (float*)d_out;
    float* ws  = (float*)d_ws;      // needs 5 * NV floats (~164 MB)

    (void)hipMemsetAsync(out, 0, (size_t)out_size * sizeof(float), stream);

    dim3 gA(10, 10, 2 * L);         // w-tiles, h-tiles, b*d slices
    lncc_wh_pass<<<gA, dim3(160), 0, stream>>>(input, target, ws);

    dim3 gB(10, 10, 2 * L);         // w-tiles, d-tiles, b*h rows
    lncc_d_finalize<<<gB, dim3(32), 0, stream>>>(ws, flow, out);
}